// Noise_19335942767177
// MI455X (gfx1250) — compile-verified
//
#include <hip/hip_runtime.h>

typedef float v2f __attribute__((ext_vector_type(2)));
typedef float v4f __attribute__((ext_vector_type(4)));
typedef float v8f __attribute__((ext_vector_type(8)));

#define DDIM 4096
#define BATCH 4
#define TMASK 64      // target qubit = bit 6 of the row index (L=32, R=64)
#define TILES_1D (DDIM / 16)      // 256
#define WAVES_PER_BLOCK 8

__global__ __launch_bounds__(256)
void noise_outer_wmma(const float* __restrict__ state,
                      const float* __restrict__ kraus,
                      float* __restrict__ out) {
    const int lane = threadIdx.x & 31;
    const int wave = threadIdx.x >> 5;
    const int tile = blockIdx.x * WAVES_PER_BLOCK + wave;
    const int ti = tile >> 8;          // row tile index, 0..255
    const int tj = tile & 255;         // col tile index, 0..255
    const int half = lane >> 4;        // 0: K=0,1 slots ; 1: K=2,3 slots (zero pad)
    const int l16  = lane & 15;

    // Kraus-transformed statevectors u_k for this lane's row (A) and column (B),
    // one value per batch element.
    v4f ur0 = {0.f, 0.f, 0.f, 0.f}, ur1 = {0.f, 0.f, 0.f, 0.f};
    v4f uc0 = {0.f, 0.f, 0.f, 0.f}, uc1 = {0.f, 0.f, 0.f, 0.f};

    if (half == 0) {
        const int gi = ti * 16 + l16;          // global row index
        const int gj = tj * 16 + l16;          // global col index
        const v4f* s4 = (const v4f*)state;     // state[i][0..3], 16B per row index

        const v4f si0 = s4[gi & ~TMASK];
        const v4f si1 = s4[gi |  TMASK];
        const v4f sj0 = s4[gj & ~TMASK];
        const v4f sj1 = s4[gj |  TMASK];

        const int ti_t = (gi >> 6) & 1;        // target-bit value of row index
        const int tj_t = (gj >> 6) & 1;

        // K[k][t][j] = kraus[k*4 + t*2 + j]
        const float a00 = kraus[ti_t * 2 + 0], a01 = kraus[ti_t * 2 + 1];
        const float a10 = kraus[4 + ti_t * 2 + 0], a11 = kraus[4 + ti_t * 2 + 1];
        ur0 = a00 * si0 + a01 * si1;           // u_0 at row gi, all batches
        ur1 = a10 * si0 + a11 * si1;           // u_1 at row gi

        const float b00 = kraus[tj_t * 2 + 0], b01 = kraus[tj_t * 2 + 1];
        const float b10 = kraus[4 + tj_t * 2 + 0], b11 = kraus[4 + tj_t * 2 + 1];
        uc0 = b00 * sj0 + b01 * sj1;           // u_0 at col gj
        uc1 = b10 * sj0 + b11 * sj1;           // u_1 at col gj
    }

    // One WMMA per batch: acc_e = A(16x4) x B(4x16), K=0..1 carry u0,u1, K=2..3 zero.
    v8f acc[BATCH];
#pragma unroll
    for (int e = 0; e < BATCH; ++e) {
        v2f a = { ur0[e], ur1[e] };
        v2f b = { uc0[e], uc1[e] };
        v8f c = {0.f, 0.f, 0.f, 0.f, 0.f, 0.f, 0.f, 0.f};
        acc[e] = __builtin_amdgcn_wmma_f32_16x16x4_f32(
            /*neg_a=*/false, a, /*neg_b=*/false, b,
            /*c_mod=*/(short)0, c, /*reuse_a=*/false, /*reuse_b=*/false);
    }

    // C/D layout: lane l holds (M = r + 8*(l/16), N = l%16) in VGPR r.
    // Output layout (D, D, B): addr = ((row*DDIM) + col)*BATCH + e -> float4 per (row,col).
    v4f* out4 = (v4f*)out;
    const int col = tj * 16 + l16;
#pragma unroll
    for (int r = 0; r < 8; ++r) {
        const int row = ti * 16 + r + 8 * half;
        v4f v = { acc[0][r], acc[1][r], acc[2][r], acc[3][r] };
        __builtin_nontemporal_store(v, &out4[row * DDIM + col]);  // 256MB stream: NT
    }
}

extern "C" void kernel_launch(void* const* d_in, const int* in_sizes, int n_in,
                              void* d_out, int out_size, void* d_ws, size_t ws_size,
                              hipStream_t stream) {
    const float* state = (const float*)d_in[0];   // (4096, 4) f32
    const float* kraus = (const float*)d_in[1];   // (2, 2, 2) f32
    float* out = (float*)d_out;                   // (4096, 4096, 4) f32

    const int total_tiles = TILES_1D * TILES_1D;              // 65536
    const int blocks = total_tiles / WAVES_PER_BLOCK;         // 8192
    noise_outer_wmma<<<blocks, 256, 0, stream>>>(state, kraus, out);
}